// AxialAttentionLastDim_77386720739737
// MI455X (gfx1250) — compile-verified
//
#include <hip/hip_runtime.h>
#include <math.h>

// CDNA5 / gfx1250, wave32. All matrix work via V_WMMA_F32_16X16X4_F32.
//
// Shapes (fixed by reference setup): B=2, C=1, W=64, H=1024, d=32, WIN=32.
// Pixels P = B*W*H = 131072, attention rows N = B*W = 128 (seq len H=1024).
//
// Workspace layout (needs 256 + 4*16MB = ~64.0 MB):
//   [0..32)        : stats  (GN1: b0{s1,s2} b1{s1,s2} ; GN2: same at +4 floats)
//   [256 + 0*BUF)  : z   (post-GN1-GELU activations [pixel][d])  -- reused as g
//   [256 + 1*BUF)  : q   [pixel][d]
//   [256 + 2*BUF)  : k   [pixel][d]
//   [256 + 3*BUF)  : v   [pixel][d]

typedef __attribute__((ext_vector_type(2))) float v2f;
typedef __attribute__((ext_vector_type(8))) float v8f;

#define H_DIM 1024
#define W_DIM 64
#define B_DIM 2
#define D_DIM 32
#define WIN   32
#define NPIX  (B_DIM * W_DIM * H_DIM)   // 131072
#define NROWS (B_DIM * W_DIM)           // 128

__device__ __forceinline__ float gelu_exact(float x) {
    return 0.5f * x * (1.0f + erff(x * 0.70710678118654752440f));
}

// D = A(16x4) * B(4x16) + C, fp32 throughout.
// A layout: lanes 0-15 hold row m, k={0,1}; lanes 16-31 hold row m, k={2,3}.
// B layout: lanes 0-15 hold col n, k={0,1}; lanes 16-31 hold col n, k={2,3}.
// C/D layout: lane&15 = col n; vgpr r = row r (lanes 0-15) / r+8 (lanes 16-31).
__device__ __forceinline__ v8f wmma4(v2f a, v2f b, v8f c) {
    return __builtin_amdgcn_wmma_f32_16x16x4_f32(false, a, false, b, (short)0, c,
                                                 false, false);
}

// ---------------------------------------------------------------- stats init
__global__ void k_init_stats(float* stats) {
    if (threadIdx.x < 8) stats[threadIdx.x] = 0.0f;
}

// ------------------------------------------------- GN1 stats (feat = x * w_in)
__global__ void k_stats1(const float* __restrict__ x, const float* __restrict__ w_in,
                         float* __restrict__ stats) {
    __shared__ float r1[256], r2[256];
    int p = blockIdx.x * 256 + threadIdx.x;
    float xv = x[p];
    float s1 = 0.f, s2 = 0.f;
#pragma unroll
    for (int d = 0; d < D_DIM; ++d) {
        float f = xv * w_in[d];
        s1 += f;
        s2 += f * f;
    }
    r1[threadIdx.x] = s1;
    r2[threadIdx.x] = s2;
    __syncthreads();
    for (int off = 128; off > 0; off >>= 1) {
        if ((int)threadIdx.x < off) {
            r1[threadIdx.x] += r1[threadIdx.x + off];
            r2[threadIdx.x] += r2[threadIdx.x + off];
        }
        __syncthreads();
    }
    if (threadIdx.x == 0) {
        int b = p >> 16;  // 65536 pixels per batch; blocks never straddle batches
        atomicAdd(&stats[b * 2 + 0], r1[0]);
        atomicAdd(&stats[b * 2 + 1], r2[0]);
    }
}

// ------------------------------------------------------- z = gelu(GN1(feat))
__global__ void k_z(const float* __restrict__ x, const float* __restrict__ w_in,
                    const float* __restrict__ gn_w, const float* __restrict__ gn_b,
                    const float* __restrict__ stats, float* __restrict__ z) {
    int p = blockIdx.x * 256 + threadIdx.x;
    int b = p >> 16;
    const float cnt = (float)(D_DIM * W_DIM * H_DIM);
    float mu = stats[b * 2 + 0] / cnt;
    float var = stats[b * 2 + 1] / cnt - mu * mu;
    float rs = rsqrtf(var + 1e-5f);
    float xv = x[p];
#pragma unroll
    for (int d = 0; d < D_DIM; ++d) {
        float f = xv * w_in[d];
        float hh = (f - mu) * rs * gn_w[d] + gn_b[d];
        z[p * D_DIM + d] = gelu_exact(hh);
    }
}

// --------------------------------------------- q/k/v projection (WMMA, fp32)
// One wave per 16 pixels: q[i,e] = sum_d z[i,d] * w_q[e,d]  (same for k,v).
__global__ void k_qkv(const float* __restrict__ z, const float* __restrict__ wq,
                      const float* __restrict__ wk, const float* __restrict__ wv,
                      float* __restrict__ q, float* __restrict__ k,
                      float* __restrict__ v) {
    int lane = threadIdx.x;
    int m = lane & 15, hi = lane >> 4;
    int pb = blockIdx.x * 16;

    v2f a[8];  // A-fragments of z tile (16 x 32)
#pragma unroll
    for (int kk = 0; kk < 8; ++kk) {
        const float* src = z + (size_t)(pb + m) * D_DIM + kk * 4 + hi * 2;
        a[kk].x = src[0];
        a[kk].y = src[1];
    }

    const float* wgt[3] = {wq, wk, wv};
    float* dst[3] = {q, k, v};
#pragma unroll
    for (int t = 0; t < 3; ++t) {
#pragma unroll
        for (int et = 0; et < 2; ++et) {
            v8f acc = {0.f, 0.f, 0.f, 0.f, 0.f, 0.f, 0.f, 0.f};
#pragma unroll
            for (int kk = 0; kk < 8; ++kk) {
                // B[k_local, e] = w[e, d]; lane holds column e = et*16 + m
                const float* wsrc = wgt[t] + (size_t)(et * 16 + m) * D_DIM + kk * 4 + hi * 2;
                v2f bf;
                bf.x = wsrc[0];
                bf.y = wsrc[1];
                acc = wmma4(a[kk], bf, acc);
            }
#pragma unroll
            for (int r = 0; r < 8; ++r) {
                dst[t][(size_t)(pb + r + 8 * hi) * D_DIM + et * 16 + m] = acc[r];
            }
        }
    }
}

// ----------------------------------------- banded attention + residual (WMMA)
// One wave per (row n, 16-query tile i0). Keys span [i0-32, i0+48) = 5 tiles.
__global__ void k_attn(const float* __restrict__ q, const float* __restrict__ kbuf,
                       const float* __restrict__ vbuf, const float* __restrict__ x,
                       const float* __restrict__ w_in,
                       const float* __restrict__ gamma_p, float* __restrict__ g,
                       float* __restrict__ stats2) {
    __shared__ float S[16][80];
    __shared__ float red1[32], red2[32];
    int lane = threadIdx.x;
    int m = lane & 15, hi = lane >> 4;
    int nrow = blockIdx.x >> 6;
    int i0 = (blockIdx.x & 63) << 4;
    int rowbase = nrow * H_DIM;

    // Q A-fragments (16 queries x 32 dims)
    v2f qa[8];
#pragma unroll
    for (int kk = 0; kk < 8; ++kk) {
        const float* src = q + (size_t)(rowbase + i0 + m) * D_DIM + kk * 4 + hi * 2;
        qa[kk].x = src[0];
        qa[kk].y = src[1];
    }

    // S = Q * K^T, 5 key tiles of 16, K-dim = 32 (8 WMMAs each)
#pragma unroll
    for (int jt = 0; jt < 5; ++jt) {
        int jg = i0 - WIN + jt * 16 + m;           // key column for this lane
        int jc = min(max(jg, 0), H_DIM - 1);       // clamp; masked later
        v8f acc = {0.f, 0.f, 0.f, 0.f, 0.f, 0.f, 0.f, 0.f};
#pragma unroll
        for (int kk = 0; kk < 8; ++kk) {
            const float* src = kbuf + (size_t)(rowbase + jc) * D_DIM + kk * 4 + hi * 2;
            v2f bf;
            bf.x = src[0];
            bf.y = src[1];
            acc = wmma4(qa[kk], bf, acc);
        }
#pragma unroll
        for (int r = 0; r < 8; ++r) S[r + 8 * hi][jt * 16 + m] = acc[r];
    }
    __syncthreads();

    // Banded softmax: lane i handles query row i (80 candidate keys)
    if (lane < 16) {
        const float scale = 0.17677669529663687f;  // 32^-0.5
        int i = lane;
        float mx = -3.0e38f;
        for (int j = 0; j < 80; ++j) {
            int jg = i0 - WIN + j;
            int di = i + WIN - j;  // == (i0+i) - jg
            if (jg >= 0 && jg < H_DIM && di >= -WIN && di <= WIN)
                mx = fmaxf(mx, S[i][j] * scale);
        }
        float sum = 0.f;
        for (int j = 0; j < 80; ++j) {
            int jg = i0 - WIN + j;
            int di = i + WIN - j;
            float pv = 0.f;
            if (jg >= 0 && jg < H_DIM && di >= -WIN && di <= WIN) {
                pv = expf(S[i][j] * scale - mx);
                sum += pv;
            }
            S[i][j] = pv;
        }
        float rinv = 1.0f / sum;
        for (int j = 0; j < 80; ++j) S[i][j] *= rinv;
    }
    __syncthreads();

    // O = P * V  (K=80 -> 20 chunks), fused with residual feat + gamma*O
    float gma = gamma_p[0];
    float ls1 = 0.f, ls2 = 0.f;
#pragma unroll
    for (int et = 0; et < 2; ++et) {
        v8f acc = {0.f, 0.f, 0.f, 0.f, 0.f, 0.f, 0.f, 0.f};
#pragma unroll
        for (int ck = 0; ck < 20; ++ck) {
            v2f pa;  // A-fragment of P from LDS: row m, k = ck*4 + hi*2 + {0,1}
            pa.x = S[m][ck * 4 + hi * 2 + 0];
            pa.y = S[m][ck * 4 + hi * 2 + 1];
            int j0 = i0 - WIN + ck * 4 + hi * 2;
            int ja = min(max(j0, 0), H_DIM - 1);
            int jb = min(max(j0 + 1, 0), H_DIM - 1);
            v2f bf;  // B-fragment of V: column e = et*16 + m (coalesced per k-row)
            bf.x = vbuf[(size_t)(rowbase + ja) * D_DIM + et * 16 + m];
            bf.y = vbuf[(size_t)(rowbase + jb) * D_DIM + et * 16 + m];
            acc = wmma4(pa, bf, acc);
        }
#pragma unroll
        for (int r = 0; r < 8; ++r) {
            int i = r + 8 * hi;
            int p = rowbase + i0 + i;
            int e = et * 16 + m;
            float f = x[p] * w_in[e];  // feat is rank-1 (C==1): recompute
            float gv = f + gma * acc[r];
            g[(size_t)p * D_DIM + e] = gv;
            ls1 += gv;
            ls2 += gv * gv;
        }
    }
    // GN2 partial sums: wave reduce via LDS, one atomic per block
    red1[lane] = ls1;
    red2[lane] = ls2;
    __syncthreads();
    if (lane == 0) {
        float s1 = 0.f, s2 = 0.f;
        for (int t = 0; t < 32; ++t) {
            s1 += red1[t];
            s2 += red2[t];
        }
        int b = nrow >> 6;
        atomicAdd(&stats2[b * 2 + 0], s1);
        atomicAdd(&stats2[b * 2 + 1], s2);
    }
}

// ------------------------------------- out = w_out . gelu(GN2(g))  (C == 1)
__global__ void k_final(const float* __restrict__ g, const float* __restrict__ gn_w,
                        const float* __restrict__ gn_b,
                        const float* __restrict__ w_out,
                        const float* __restrict__ stats2, float* __restrict__ out) {
    int p = blockIdx.x * 256 + threadIdx.x;
    int b = p >> 16;
    const float cnt = (float)(D_DIM * W_DIM * H_DIM);
    float mu = stats2[b * 2 + 0] / cnt;
    float var = stats2[b * 2 + 1] / cnt - mu * mu;
    float rs = rsqrtf(var + 1e-5f);
    float acc = 0.f;
#pragma unroll
    for (int d = 0; d < D_DIM; ++d) {
        float hh = (g[(size_t)p * D_DIM + d] - mu) * rs * gn_w[d] + gn_b[d];
        acc += gelu_exact(hh) * w_out[d];
    }
    out[p] = acc;
}

extern "C" void kernel_launch(void* const* d_in, const int* in_sizes, int n_in,
                              void* d_out, int out_size, void* d_ws, size_t ws_size,
                              hipStream_t stream) {
    (void)in_sizes; (void)n_in; (void)out_size; (void)ws_size;
    const float* x     = (const float*)d_in[0];
    const float* w_in  = (const float*)d_in[1];
    const float* w_q   = (const float*)d_in[2];
    const float* w_k   = (const float*)d_in[3];
    const float* w_v   = (const float*)d_in[4];
    const float* w_out = (const float*)d_in[5];
    const float* gn_w  = (const float*)d_in[6];
    const float* gn_b  = (const float*)d_in[7];
    const float* gamma = (const float*)d_in[8];
    float* out = (float*)d_out;

    char* ws = (char*)d_ws;
    float* stats = (float*)ws;  // 8 floats: GN1 pair then GN2 pair
    const size_t BUF = (size_t)NPIX * D_DIM * sizeof(float);  // 16 MB
    float* z  = (float*)(ws + 256);            // reused as g after k_qkv
    float* qb = (float*)(ws + 256 + 1 * BUF);
    float* kb = (float*)(ws + 256 + 2 * BUF);
    float* vb = (float*)(ws + 256 + 3 * BUF);
    float* g  = z;

    k_init_stats<<<1, 32, 0, stream>>>(stats);
    k_stats1<<<NPIX / 256, 256, 0, stream>>>(x, w_in, stats);
    k_z<<<NPIX / 256, 256, 0, stream>>>(x, w_in, gn_w, gn_b, stats, z);
    k_qkv<<<NPIX / 16, 32, 0, stream>>>(z, w_q, w_k, w_v, qb, kb, vb);
    k_attn<<<NROWS * (H_DIM / 16), 32, 0, stream>>>(qb, kb, vb, x, w_in, gamma, g,
                                                    stats + 4);
    k_final<<<NPIX / 256, 256, 0, stream>>>(g, gn_w, gn_b, w_out, stats + 4, out);
}